// DCTBlur_13469017440442
// MI455X (gfx1250) — compile-verified
//
#include <hip/hip_runtime.h>
#include <math.h>

typedef float v2f __attribute__((ext_vector_type(2)));
typedef float v8f __attribute__((ext_vector_type(8)));

#define NN 512
#define NTILES (NN / 16)        // 32 row strips
#define WAVES 8
#define TPW 4                   // tiles per wave: 8 waves * 4 * 16 = 512 cols
#define LDSTRIDE 516            // 512 + 4 pad -> conflict-free ds_load_b64 A-frag reads

// ---------------------------------------------------------------------------
// Build orthonormal DCT-II matrix D[k][n] = s_k * cos(pi*(n+0.5)*k/N) and its
// transpose. Angle reduced exactly: (2n+1)*k mod 2048, scaled by pi/1024.
// ---------------------------------------------------------------------------
__global__ __launch_bounds__(512) void build_D(float* __restrict__ D,
                                               float* __restrict__ Dt) {
    int k = blockIdx.x;
    int n = threadIdx.x;
    float s = (k == 0) ? 0.04419417382415922f   // sqrt(1/512)
                       : 0.0625f;               // sqrt(2/512)
    int m = ((2 * n + 1) * k) & 2047;           // exact mod 2048 (fits in int)
    float v = s * cosf((float)m * (3.14159265358979323846f / 1024.0f));
    D[k * NN + n]  = v;
    Dt[n * NN + k] = v;
}

// ---------------------------------------------------------------------------
// Pass 1: T = D * X  (per image). Block owns one 16-row output strip.
// A-panel (16x512 rows of D) staged in LDS; B streamed from X (L2 resident).
// ---------------------------------------------------------------------------
__global__ __launch_bounds__(256) void dct_rows(const float* __restrict__ X,
                                                const float* __restrict__ D,
                                                float* __restrict__ T) {
    __shared__ float Alds[16 * LDSTRIDE];

    const int img  = blockIdx.y;
    const int m0   = blockIdx.x * 16;
    const int tid  = threadIdx.x;
    const int lane = tid & 31;
    const int wv   = tid >> 5;
    const int half = lane >> 4;   // 0: lanes 0-15, 1: lanes 16-31
    const int l16  = lane & 15;

    // Stage A = D[m0:m0+16][0:512] into LDS with padded row stride.
    {
        const float4* src = (const float4*)(D + (size_t)m0 * NN);
        for (int i = tid; i < 16 * (NN / 4); i += 256) {
            int row = i >> 7, c4 = i & 127;
            *(float4*)&Alds[row * LDSTRIDE + c4 * 4] = src[(size_t)row * (NN / 4) + c4];
        }
    }
    __syncthreads();

    const float* Xi = X + (size_t)img * NN * NN;

    v8f acc[TPW] = {};

    for (int k0 = 0; k0 < NN; k0 += 4) {
        // A fragment: lane l16 holds row M=l16, K = k0+2*half, k0+2*half+1
        v2f a = *(const v2f*)&Alds[l16 * LDSTRIDE + k0 + 2 * half];
        const int krow = k0 + 2 * half;
        const float* brow = Xi + (size_t)krow * NN + l16;
#pragma unroll
        for (int j = 0; j < TPW; ++j) {
            int n0 = wv * (TPW * 16) + j * 16;
            v2f b;
            b.x = brow[n0];        // B[krow][n0+l16]
            b.y = brow[n0 + NN];   // B[krow+1][n0+l16]
            acc[j] = __builtin_amdgcn_wmma_f32_16x16x4_f32(
                false, a, false, b, (short)0, acc[j], false, false);
        }
    }

    float* To = T + (size_t)img * NN * NN;
#pragma unroll
    for (int j = 0; j < TPW; ++j) {
        int n0 = wv * (TPW * 16) + j * 16;
#pragma unroll
        for (int r = 0; r < 8; ++r) {
            int row = m0 + r + 8 * half;
            To[(size_t)row * NN + n0 + l16] = acc[j][r];
        }
    }
}

// ---------------------------------------------------------------------------
// Pass 2 (in place): Y = (T * D^T) .* damp. Block stages its full 16x512
// input strip into LDS before computing, so in-place global writes are safe.
// Fused epilogue: damp = exp(-(f_k^2 + f_l^2) * tt[b]), tt = 40^(2t)/8.
// ---------------------------------------------------------------------------
__global__ __launch_bounds__(256) void dct_cols_damp(float* __restrict__ Y,
                                                     const float* __restrict__ Dt,
                                                     const float* __restrict__ t,
                                                     int C) {
    __shared__ float Alds[16 * LDSTRIDE];

    const int img  = blockIdx.y;
    const int b    = img / C;
    const int m0   = blockIdx.x * 16;
    const int tid  = threadIdx.x;
    const int lane = tid & 31;
    const int wv   = tid >> 5;
    const int half = lane >> 4;
    const int l16  = lane & 15;

    float* Yi = Y + (size_t)img * NN * NN;

    // Stage the 16x512 input strip (contiguous rows) into LDS.
    {
        const float4* src = (const float4*)(Yi + (size_t)m0 * NN);
        for (int i = tid; i < 16 * (NN / 4); i += 256) {
            int row = i >> 7, c4 = i & 127;
            *(float4*)&Alds[row * LDSTRIDE + c4 * 4] = src[(size_t)row * (NN / 4) + c4];
        }
    }
    __syncthreads();

    v8f acc[TPW] = {};

    for (int k0 = 0; k0 < NN; k0 += 4) {
        v2f a = *(const v2f*)&Alds[l16 * LDSTRIDE + k0 + 2 * half];
        const int krow = k0 + 2 * half;
        const float* brow = Dt + (size_t)krow * NN + l16;  // B[w][l] = D[l][w]
#pragma unroll
        for (int j = 0; j < TPW; ++j) {
            int n0 = wv * (TPW * 16) + j * 16;
            v2f bf;
            bf.x = brow[n0];
            bf.y = brow[n0 + NN];
            acc[j] = __builtin_amdgcn_wmma_f32_16x16x4_f32(
                false, a, false, bf, (short)0, acc[j], false, false);
        }
    }

    // Fused blur-schedule epilogue + in-place store.
    const float tb = t[b];
    const float tt = expf(7.377758908227873f * tb) * 0.125f;  // 40^(2t)/8
    const float fc = 3.14159265358979323846f / (float)NN;

#pragma unroll
    for (int j = 0; j < TPW; ++j) {
        int n0 = wv * (TPW * 16) + j * 16;
        float fl = fc * (float)(n0 + l16);
        float fl2 = fl * fl;
#pragma unroll
        for (int r = 0; r < 8; ++r) {
            int row = m0 + r + 8 * half;
            float fk = fc * (float)row;
            float damp = __expf(-(fk * fk + fl2) * tt);
            Yi[(size_t)row * NN + n0 + l16] = acc[j][r] * damp;
        }
    }
}

// ---------------------------------------------------------------------------
extern "C" void kernel_launch(void* const* d_in, const int* in_sizes, int n_in,
                              void* d_out, int out_size, void* d_ws, size_t ws_size,
                              hipStream_t stream) {
    const float* x = (const float*)d_in[0];
    const float* t = (const float*)d_in[1];
    float* out = (float*)d_out;

    float* D  = (float*)d_ws;       // NN*NN floats (1 MB)
    float* Dt = D + NN * NN;        // NN*NN floats (1 MB)

    const int images = in_sizes[0] / (NN * NN);   // 192
    const int B      = in_sizes[1];               // 64
    const int C      = images / B;                // 3

    build_D<<<dim3(NN), dim3(NN), 0, stream>>>(D, Dt);

    dim3 grid(NTILES, images);
    dct_rows<<<grid, dim3(256), 0, stream>>>(x, D, out);
    dct_cols_damp<<<grid, dim3(256), 0, stream>>>(out, Dt, t, C);
}